// DirectNBodyAcceleration_75685913690724
// MI455X (gfx1250) — compile-verified
//
#include <hip/hip_runtime.h>

// DirectNBodyAcceleration on MI455X (gfx1250):
//   acc[j] = G * m[j] * sum_i (q_i - q_j) * (||q_i - q_j||^2 + eps)^-1.5
// r^2 = (d2_j + eps) - 2*q_j.q_i + d2_i  -> rank-4 bilinear part computed with
// V_WMMA_F32_16X16X4_F32 (j on M axis, i on N/lane axis), bare v_rsq_f32 +
// fma accumulation on VALU, final i-reduction via 16-lane shfl_xor butterfly.

typedef float v2f __attribute__((ext_vector_type(2)));
typedef float v8f __attribute__((ext_vector_type(8)));

#define NBODY_EPS 1e-4f   // softening_length^2 = 0.01^2

__global__ __launch_bounds__(64) void nbody_wmma_f32_kernel(
    const float* __restrict__ pos,    // [n,3]
    const float* __restrict__ mass,   // [n]
    const float* __restrict__ Gptr,   // [1]
    float* __restrict__ acc,          // [n,3]
    int n)                            // n assumed multiple of 16 (n = 8192)
{
  const int lane = threadIdx.x & 31;
  const int l16  = lane & 15;
  const int half = lane >> 4;                       // 0: lanes 0-15, 1: lanes 16-31
  const int wavesPerBlock = blockDim.x >> 5;
  const int wave  = blockIdx.x * wavesPerBlock + (threadIdx.x >> 5);
  const int jBase = wave * 16;                      // this wave owns j in [jBase, jBase+16)
  if (jBase >= n) return;

  const float G = Gptr[0];

  // ---- Build A matrix (16x4 f32, rows = j), loop-invariant ----
  // A[j] = [-2x_j, -2y_j, -2z_j, d2_j + eps]
  // A VGPR layout: lane l: m = l%16; VGPR0 = K(l<16?0:2), VGPR1 = K(l<16?1:3)
  int j0 = jBase + l16;
  if (j0 >= n) j0 = n - 1;                          // clamp (defensive; n%16==0)
  const float xj = pos[3*j0+0];
  const float yj = pos[3*j0+1];
  const float zj = pos[3*j0+2];
  const float d2j = fmaf(xj, xj, fmaf(yj, yj, zj*zj));
  v2f a;
  a.x = half ? (-2.0f*zj)        : (-2.0f*xj);
  a.y = half ? (d2j + NBODY_EPS) : (-2.0f*yj);

  // Per-lane accumulators: index r -> row m = r + 8*half -> j = jBase + r + 8*half.
  // Each lane holds the partial sum over its i-class (i ≡ l16 mod 16).
  v8f sx = {0.f,0.f,0.f,0.f,0.f,0.f,0.f,0.f};
  v8f sy = {0.f,0.f,0.f,0.f,0.f,0.f,0.f,0.f};
  v8f sz = {0.f,0.f,0.f,0.f,0.f,0.f,0.f,0.f};
  v8f sw = {0.f,0.f,0.f,0.f,0.f,0.f,0.f,0.f};

#pragma unroll 2
  for (int t = 0; t < n; t += 16) {
    const int i = t + l16;                          // this lane's i within the tile
    const float xi = pos[3*i+0];
    const float yi = pos[3*i+1];
    const float zi = pos[3*i+2];
    // prefetch next i-tile (global_prefetch_b8); clamp address, no branch
    const int ipf = (t + 16 < n) ? (i + 16) : i;
    __builtin_prefetch(&pos[3*ipf], 0, 3);
    const float d2i = fmaf(xi, xi, fmaf(yi, yi, zi*zi));

    // B matrix (4x16 f32, cols = i): B[:,i] = [x_i, y_i, z_i, 1]
    // B VGPR layout: lane l: n = l%16; VGPR0 = row(l<16?0:2), VGPR1 = row(l<16?1:3)
    v2f b;
    b.x = half ? zi   : xi;
    b.y = half ? 1.0f : yi;

    v8f c = {0.f,0.f,0.f,0.f,0.f,0.f,0.f,0.f};
    // c[r] (lane l) = d2_j + eps - 2*q_j.q_i   with j = jBase + r + 8*half, i = t + l%16
    c = __builtin_amdgcn_wmma_f32_16x16x4_f32(
        /*neg_a=*/false, a, /*neg_b=*/false, b,
        /*c_mod=*/(short)0, c, /*reuse_a=*/false, /*reuse_b=*/false);

#pragma unroll
    for (int r = 0; r < 8; ++r) {
      const float r2   = c[r] + d2i;                // full softened squared distance
      // r2 >= eps = 1e-4: never denormal, so use the bare hardware v_rsq_f32
      // (avoids clang's denormal pre/post-scale cmp+cndmask sequence).
      const float inv  = __builtin_amdgcn_rsqf(r2);
      const float inv3 = inv * inv * inv;           // r^-3
      sw[r] += inv3;
      sx[r]  = fmaf(inv3, xi, sx[r]);
      sy[r]  = fmaf(inv3, yi, sy[r]);
      sz[r]  = fmaf(inv3, zi, sz[r]);
    }
  }

  // ---- Reduce over the 16 i-classes (butterfly within each 16-lane half) ----
#pragma unroll
  for (int m = 1; m <= 8; m <<= 1) {
#pragma unroll
    for (int r = 0; r < 8; ++r) {
      sx[r] += __shfl_xor(sx[r], m, 32);
      sy[r] += __shfl_xor(sy[r], m, 32);
      sz[r] += __shfl_xor(sz[r], m, 32);
      sw[r] += __shfl_xor(sw[r], m, 32);
    }
  }

  // ---- Epilogue: lane 0 of each half writes its 8 rows ----
  if (l16 == 0) {
#pragma unroll
    for (int r = 0; r < 8; ++r) {
      const int jr = jBase + r + 8*half;
      if (jr < n) {
        const float gm = G * mass[jr];
        const float qx = pos[3*jr+0];
        const float qy = pos[3*jr+1];
        const float qz = pos[3*jr+2];
        acc[3*jr+0] = gm * (sx[r] - qx * sw[r]);
        acc[3*jr+1] = gm * (sy[r] - qy * sw[r]);
        acc[3*jr+2] = gm * (sz[r] - qz * sw[r]);
      }
    }
  }
}

extern "C" void kernel_launch(void* const* d_in, const int* in_sizes, int n_in,
                              void* d_out, int out_size, void* d_ws, size_t ws_size,
                              hipStream_t stream) {
  // setup_inputs() order: t (1), y (N*3), masses (N), G (1)
  const float* pos    = (const float*)d_in[1];
  const float* masses = (const float*)d_in[2];
  const float* Gptr   = (const float*)d_in[3];
  float* out          = (float*)d_out;

  const int n = in_sizes[2];                        // N = masses element count

  const int blockThreads  = 64;                     // 2 waves per block (wave32)
  const int wavesTotal    = (n + 15) / 16;          // one wave per 16 target bodies
  const int wavesPerBlock = blockThreads / 32;
  const int grid          = (wavesTotal + wavesPerBlock - 1) / wavesPerBlock;

  nbody_wmma_f32_kernel<<<grid, blockThreads, 0, stream>>>(pos, masses, Gptr, out, n);
}